// SparseMoELayer_67319317397650
// MI455X (gfx1250) — compile-verified
//
#include <hip/hip_runtime.h>
#include <hip/hip_bf16.h>

#define B_ 8
#define K_ 8
#define C_ 256
#define L_ 4096
#define TOPK_ 4

// Block tile: 128 (d) x 128 (l). 8 waves, each wave 128(d) x 16(l) -> 8 WMMA
// accumulator tiles; one B fragment feeds 8 WMMAs.
#define BM 128
#define BN 128
// A is staged per c-chunk of 64 into LDS, row stride 68 floats:
// 68 mod 64 == 4 -> ds_load_b64 bank map {4*m16 + 2*half + c, +1} covers all
// 64 banks conflict-free.
#define CC 64
#define ASTR (CC + 4)
#define NCHUNK (C_ / CC)

typedef float v2f __attribute__((ext_vector_type(2)));
typedef float v8f __attribute__((ext_vector_type(8)));
typedef int   v4i __attribute__((ext_vector_type(4)));
typedef __attribute__((address_space(1))) v4i gv4i;  // global 16B vector
typedef __attribute__((address_space(3))) v4i lv4i;  // LDS 16B vector

__device__ __forceinline__ void async_copy16(const float* gsrc, float* ldst) {
#if __has_builtin(__builtin_amdgcn_global_load_async_to_lds_b128)
  __builtin_amdgcn_global_load_async_to_lds_b128(
      (gv4i*)(uintptr_t)gsrc, (lv4i*)(unsigned int)(size_t)ldst, 0, 0);
#else
  *(float4*)ldst = *(const float4*)gsrc;
#endif
}

template <int N>
__device__ __forceinline__ void async_wait() {
#if __has_builtin(__builtin_amdgcn_global_load_async_to_lds_b128)
#if __has_builtin(__builtin_amdgcn_s_wait_asynccnt)
  __builtin_amdgcn_s_wait_asynccnt(N);
#else
  asm volatile("s_wait_asynccnt %0" ::"i"(N) : "memory");
#endif
#endif
}

// ---------------------------------------------------------------------------
// Kernel 1: routing (top-4 + renormalize) and combined-bias precompute.
// ---------------------------------------------------------------------------
__global__ void moe_routing_kernel(const float* __restrict__ scores,
                                   const float* __restrict__ expert_b,
                                   float* __restrict__ wfull,
                                   float* __restrict__ cbias) {
  __shared__ float s_w[B_ * K_];
  const int t = threadIdx.x;

  if (t < B_) {
    float s[K_];
    bool used[K_];
    float wloc[K_];
#pragma unroll
    for (int k = 0; k < K_; ++k) {
      s[k] = scores[t * K_ + k];
      used[k] = false;
      wloc[k] = 0.0f;
    }
    int sel[TOPK_];
    float tw[TOPK_];
    float sum = 0.0f;
#pragma unroll
    for (int i = 0; i < TOPK_; ++i) {
      int best = 0;
      float bv = -3.4e38f;
#pragma unroll
      for (int k = 0; k < K_; ++k) {
        if (!used[k] && s[k] > bv) { bv = s[k]; best = k; }
      }
      used[best] = true;
      sel[i] = best;
      tw[i] = bv;
      sum += bv;
    }
    const float inv = 1.0f / (sum + 1e-8f);
#pragma unroll
    for (int i = 0; i < TOPK_; ++i) wloc[sel[i]] = tw[i] * inv;
#pragma unroll
    for (int k = 0; k < K_; ++k) {
      s_w[t * K_ + k] = wloc[k];
      wfull[t * K_ + k] = wloc[k];
    }
  }
  __syncthreads();

  const int d = t;  // blockDim.x == C_ == 256
#pragma unroll
  for (int b = 0; b < B_; ++b) {
    float acc = 0.0f;
#pragma unroll
    for (int k = 0; k < K_; ++k) acc += s_w[b * K_ + k] * expert_b[k * C_ + d];
    cbias[b * C_ + d] = acc;
  }
}

// ---------------------------------------------------------------------------
// Kernel 2: f32 WMMA GEMM per selected expert. A chunks double-buffered in
// LDS via gfx1250 async global->LDS copies; tall 128x16 wave tile gives 8x
// B-operand reuse. Routing weight folded into the single B fragment
// (1 v_pk_mul_f32 per 8 WMMAs).
// ---------------------------------------------------------------------------
__global__ __launch_bounds__(256, 2)
void moe_wmma_kernel(const float* __restrict__ xs,
                     const float* __restrict__ expert_w,
                     const float* __restrict__ wfull,
                     const float* __restrict__ cbias,
                     float* __restrict__ out) {
  __shared__ float sA[2][BM * ASTR];  // 2 x 34,816 B

  const int tid  = threadIdx.x;
  const int lane = tid & 31;
  const int wave = tid >> 5;          // 0..7 -> 16-wide l-slice
  const int half = lane >> 4;
  const int m16  = lane & 15;

  const int b      = blockIdx.z;
  const int d_base = blockIdx.y * BM;
  const int l_base = blockIdx.x * BN + wave * 16;

  // Staging map: 2 threads per d-row, 8 x 16B each (row = tid>>1, 64 floats).
  const int srow  = tid >> 1;
  const int scol0 = (tid & 1) * 4;

  v8f acc[8] = {};

#pragma unroll 1
  for (int k = 0; k < K_; ++k) {
    const float w = wfull[b * K_ + k];
    if (w == 0.0f) continue;  // uniform across block (same b)

    const float* __restrict__ Wk = expert_w + (size_t)k * C_ * C_;         // [d][c]
    const float* __restrict__ Xk = xs + ((size_t)(b * K_ + k)) * C_ * L_;  // [c][l]

    // Issue chunk 0 copies (previous consumers already passed the trailing
    // barrier of the last chunk, so buffer 0 is free).
    {
      const float* g0 = Wk + (size_t)(d_base + srow) * C_ + scol0;
      float* l0 = &sA[0][srow * ASTR + scol0];
#pragma unroll
      for (int i = 0; i < 8; ++i) async_copy16(g0 + 8 * i, l0 + 8 * i);
    }

#pragma unroll 1
    for (int ch = 0; ch < NCHUNK; ++ch) {
      if (ch + 1 < NCHUNK) {  // prefetch next chunk into the other buffer
        const float* gn =
            Wk + (size_t)(d_base + srow) * C_ + (ch + 1) * CC + scol0;
        float* ln = &sA[(ch + 1) & 1][srow * ASTR + scol0];
#pragma unroll
        for (int i = 0; i < 8; ++i) async_copy16(gn + 8 * i, ln + 8 * i);
        async_wait<8>();  // oldest 8 (current chunk) done, 8 still in flight
      } else {
        async_wait<0>();
      }
      __syncthreads();  // current chunk visible to all waves

      const float* sAb  = sA[ch & 1];
      const float* Bcol = Xk + (size_t)(ch * CC + 2 * half) * L_ + l_base + m16;

#pragma unroll 4
      for (int c = 0; c < CC; c += 4) {
        const float* bp = Bcol + (size_t)c * L_;
        v2f bb;
        bb.x = bp[0];
        bb.y = bp[L_];
        bb *= w;  // fold routing weight into the shared B fragment (1 pk_mul)

#pragma unroll
        for (int ti = 0; ti < 8; ++ti) {
          v2f a = *(const v2f*)&sAb[(m16 + 16 * ti) * ASTR + 2 * half + c];
          acc[ti] = __builtin_amdgcn_wmma_f32_16x16x4_f32(
              false, a, false, bb, (short)0, acc[ti], false, false);
        }
      }
      __syncthreads();  // all waves done reading before buffer is rewritten
    }
  }

  // Epilogue: C/D layout — VGPR j holds M=j (lanes 0-15) / M=j+8 (lanes 16-31).
  const int lcol = l_base + m16;
  float* op = out + (size_t)b * C_ * L_ + lcol;
#pragma unroll
  for (int ti = 0; ti < 8; ++ti) {
#pragma unroll
    for (int j = 0; j < 8; ++j) {
      const int drow = d_base + ti * 16 + half * 8 + j;
      __builtin_nontemporal_store(acc[ti][j] + cbias[b * C_ + drow],
                                  op + (size_t)drow * L_);
    }
  }
}

extern "C" void kernel_launch(void* const* d_in, const int* in_sizes, int n_in,
                              void* d_out, int out_size, void* d_ws, size_t ws_size,
                              hipStream_t stream) {
  const float* xs       = (const float*)d_in[0];  // [B,K,C,L]
  const float* scores   = (const float*)d_in[1];  // [B,K]
  const float* expert_w = (const float*)d_in[2];  // [K,C,C]
  const float* expert_b = (const float*)d_in[3];  // [K,C]
  float* out = (float*)d_out;                     // [B,C,L]

  float* wfull = (float*)d_ws;        // B*K floats
  float* cbias = wfull + B_ * K_;     // B*C floats

  moe_routing_kernel<<<1, 256, 0, stream>>>(scores, expert_b, wfull, cbias);

  dim3 grid(L_ / BN, C_ / BM, B_);
  moe_wmma_kernel<<<grid, 256, 0, stream>>>(xs, expert_w, wfull, cbias, out);
}